// VectorQuantizer_57148834841139
// MI455X (gfx1250) — compile-verified
//
#include <hip/hip_runtime.h>
#include <math.h>

// ---------------------------------------------------------------------------
// VectorQuantizer for MI455X (gfx1250, wave32).
//
// Roofline: 8.6 GFLOP of GEMM vs ~340 MB of HBM traffic -> bound by the (N,K)
// proximity write stream. fp32 WMMA (V_WMMA_F32_16X16X4_F32) for the GEMM;
// the 268 MB proximity buffer exceeds the
// 192 MB L2, so all streaming outputs use NT temporal hints to preserve L2
// residency for the hot 128 KB `centers` table re-read by all 8192 blocks.
// ---------------------------------------------------------------------------

typedef __attribute__((ext_vector_type(2))) float v2f;
typedef __attribute__((ext_vector_type(4))) float v4f;   // native vector: OK for NT builtins
typedef __attribute__((ext_vector_type(8))) float v8f;

#define Bdim   32
#define Cdim   64      // D
#define Hdim   64
#define Wdim   64
#define Kc     512
#define HWp    (Hdim * Wdim)          // 4096
#define Npts   (Bdim * HWp)           // 131072
#define ROWS   16
#define THREADS 256

// ---------------------------------------------------------------------------
// Kernel 1: center norms + zero histogram (workspace is poisoned, not zeroed).
// ---------------------------------------------------------------------------
__global__ __launch_bounds__(256)
void vq_init(const float* __restrict__ centers,
             float* __restrict__ cnorm,
             float* __restrict__ counts)
{
    int k = blockIdx.x * blockDim.x + threadIdx.x;
    if (k < Kc) {
        const float* c = centers + (size_t)k * Cdim;
        float s = 0.0f;
        #pragma unroll 8
        for (int i = 0; i < Cdim; ++i) s += c[i] * c[i];
        cnorm[k]  = s;
        counts[k] = 0.0f;
    }
}

// ---------------------------------------------------------------------------
// Kernel 2: per 16-row block -> distances via fp32 WMMA, softmax, argmin,
//           NT-streamed proximity rows, qz scatter, histogram atomics.
// ---------------------------------------------------------------------------
__global__ __launch_bounds__(THREADS)
void vq_main(const float* __restrict__ z,
             const float* __restrict__ centers,
             const float* __restrict__ cnorm,
             float* __restrict__ counts,
             float* __restrict__ qz_out,     // (B,C,H,W)
             float* __restrict__ idx_out,    // (N,1) as float
             float* __restrict__ prox_out)   // (N,K)
{
    // dist padded to 516: rows r and r+8 differ by 8*516 % 64 = 32 banks,
    // so the two lane-halves of a WMMA tile store conflict-free.
    __shared__ float zs[ROWS][Cdim + 4];
    __shared__ float dist[ROWS][Kc + 4];
    __shared__ float znorm[ROWS];
    __shared__ int   idxs[ROWS];

    const int tid  = threadIdx.x;
    const int lane = tid & 31;
    const int wave = tid >> 5;

    const int row0 = blockIdx.x * ROWS;       // flat n of first row
    const int b    = row0 / HWp;              // batch index (16 | HWp)
    const int hw0  = row0 % HWp;
    const size_t zbase = (size_t)b * Cdim * HWp + hw0;

    // ---- stage z tile: 16 rows x 64 dims. Read-once -> NT loads. ----
    for (int i = tid; i < ROWS * Cdim; i += THREADS) {
        int r = i & 15;
        int c = i >> 4;
        zs[r][c] = __builtin_nontemporal_load(z + zbase + (size_t)c * HWp + r);
    }
    __syncthreads();

    // ---- row norms ----
    if (tid < ROWS) {
        float s = 0.0f;
        #pragma unroll 8
        for (int c = 0; c < Cdim; ++c) { float v = zs[tid][c]; s += v * v; }
        znorm[tid] = s;
    }
    __syncthreads();

    // ---- GEMM tile: this wave owns 64 columns (4 x 16x16 WMMA tiles) ----
    // A (16x4 f32): lanes 0-15 hold M=lane, K = dk+{0,1}; lanes 16-31 K = dk+{2,3}
    // B (4x16 f32): lanes 0-15 hold N=lane, K = dk+{0,1}; lanes 16-31 K = dk+{2,3}
    const int rowA = lane & 15;
    const int koff = (lane < 16) ? 0 : 2;
    const int colL = lane & 15;
    const int n0   = wave * 64;

    v8f acc[4] = {};
    for (int dk = 0; dk < Cdim; dk += 4) {
        v2f a = *(const v2f*)&zs[rowA][dk + koff];
        #pragma unroll
        for (int ct = 0; ct < 4; ++ct) {
            int col = n0 + ct * 16 + colL;
            v2f bf = *(const v2f*)(centers + (size_t)col * Cdim + dk + koff);
            acc[ct] = __builtin_amdgcn_wmma_f32_16x16x4_f32(
                /*neg_a=*/false, a, /*neg_b=*/false, bf,
                /*c_mod=*/(short)0, acc[ct],
                /*reuse_a=*/false, /*reuse_b=*/false);
        }
    }

    // ---- distances into LDS: d = sqrt(max(|z|^2 - 2 z.c + |c|^2, 0)) ----
    // C/D layout: VGPR j -> M = j (lanes 0-15) / j+8 (lanes 16-31), N = lane%16
    const int rbase = (lane < 16) ? 0 : 8;
    #pragma unroll
    for (int ct = 0; ct < 4; ++ct) {
        int   col = n0 + ct * 16 + colL;
        float cnv = cnorm[col];
        #pragma unroll
        for (int j = 0; j < 8; ++j) {
            int   r  = rbase + j;
            float sq = znorm[r] - 2.0f * acc[ct][j] + cnv;
            dist[r][col] = sqrtf(fmaxf(sq, 0.0f));
        }
    }
    __syncthreads();

    // ---- per-row argmin + stable softmax: each wave reduces 2 rows.
    //      All scans vectorized: lane owns 4 consecutive cols per step
    //      (b128 LDS reads, b128 NT global stores -> 512B/wave/store). ----
    #pragma unroll
    for (int rr = 0; rr < 2; ++rr) {
        int    r = wave * 2 + rr;
        size_t n = (size_t)row0 + r;

        // pass 1: min + argmin (per-lane scan order is index-monotone, so
        // strict '<' keeps the first occurrence; cross-lane tie-break below)
        float mval = 3.402823466e38f;
        int   midx = 0;
        #pragma unroll
        for (int i = 0; i < 4; ++i) {
            int c0 = i * 128 + lane * 4;
            v4f dv = *(const v4f*)&dist[r][c0];
            if (dv.x < mval) { mval = dv.x; midx = c0 + 0; }
            if (dv.y < mval) { mval = dv.y; midx = c0 + 1; }
            if (dv.z < mval) { mval = dv.z; midx = c0 + 2; }
            if (dv.w < mval) { mval = dv.w; midx = c0 + 3; }
        }
        #pragma unroll
        for (int off = 16; off > 0; off >>= 1) {
            float om = __shfl_xor(mval, off, 32);
            int   oi = __shfl_xor(midx, off, 32);
            if (om < mval || (om == mval && oi < midx)) { mval = om; midx = oi; }
        }

        // pass 2: sum of exp(mval - d)
        float s = 0.0f;
        #pragma unroll
        for (int i = 0; i < 4; ++i) {
            int c0 = i * 128 + lane * 4;
            v4f dv = *(const v4f*)&dist[r][c0];
            s += expf(mval - dv.x) + expf(mval - dv.y)
               + expf(mval - dv.z) + expf(mval - dv.w);
        }
        #pragma unroll
        for (int off = 16; off > 0; off >>= 1) s += __shfl_xor(s, off, 32);
        float inv = 1.0f / s;

        // pass 3: NT-stream the 2 KB proximity row (dominant HBM traffic)
        #pragma unroll
        for (int i = 0; i < 4; ++i) {
            int c0 = i * 128 + lane * 4;
            v4f dv = *(const v4f*)&dist[r][c0];
            v4f e;
            e.x = expf(mval - dv.x) * inv;
            e.y = expf(mval - dv.y) * inv;
            e.z = expf(mval - dv.z) * inv;
            e.w = expf(mval - dv.w) * inv;
            __builtin_nontemporal_store(e, (v4f*)(prox_out + n * Kc + c0));
        }

        if (lane == 0) {
            idx_out[n] = (float)midx;
            idxs[r]    = midx;
            atomicAdd(&counts[midx], 1.0f);
        }
    }
    __syncthreads();

    // ---- qz scatter (straight-through fwd value == hard assignment);
    //      16 consecutive threads -> 64B contiguous segments, NT stores ----
    for (int i = tid; i < ROWS * Cdim; i += THREADS) {
        int r = i & 15;
        int c = i >> 4;
        float v = centers[(size_t)idxs[r] * Cdim + c];
        __builtin_nontemporal_store(v, qz_out + zbase + (size_t)c * HWp + r);
    }
}

// ---------------------------------------------------------------------------
// Kernel 3: perplexity from histogram; vq_loss = 0 (non-VQVAE branch).
// ---------------------------------------------------------------------------
__global__ __launch_bounds__(256)
void vq_finalize(const float* __restrict__ counts,
                 float* __restrict__ loss_out,
                 float* __restrict__ perp_out)
{
    __shared__ float red[256];
    int tid = threadIdx.x;
    float s = 0.0f;
    for (int k = tid; k < Kc; k += 256) {
        float p = counts[k] * (1.0f / (float)Npts);
        s += p * logf(p + 1e-10f);
    }
    red[tid] = s;
    __syncthreads();
    for (int off = 128; off > 0; off >>= 1) {
        if (tid < off) red[tid] += red[tid + off];
        __syncthreads();
    }
    if (tid == 0) {
        *perp_out = expf(-red[0]);
        *loss_out = 0.0f;
    }
}

// ---------------------------------------------------------------------------
extern "C" void kernel_launch(void* const* d_in, const int* in_sizes, int n_in,
                              void* d_out, int out_size, void* d_ws, size_t ws_size,
                              hipStream_t stream)
{
    const float* z       = (const float*)d_in[0];   // (32,64,64,64)
    const float* centers = (const float*)d_in[1];   // (512,64)

    float* out  = (float*)d_out;
    float* qz   = out;                              // 8388608
    float* loss = out + (size_t)8388608;            // 1
    float* perp = out + (size_t)8388609;            // 1
    float* idx  = out + (size_t)8388610;            // 131072
    float* prox = out + (size_t)8388610 + 131072;   // 67108864

    float* cnorm  = (float*)d_ws;                   // 512 floats
    float* counts = cnorm + Kc;                     // 512 floats

    vq_init    <<<2, 256, 0, stream>>>(centers, cnorm, counts);
    vq_main    <<<Npts / ROWS, THREADS, 0, stream>>>(z, centers, cnorm, counts,
                                                     qz, idx, prox);
    vq_finalize<<<1, 256, 0, stream>>>(counts, loss, perp);
}